// TextCNN_54408645706069
// MI455X (gfx1250) — compile-verified
//
#include <hip/hip_runtime.h>
#include <math.h>

typedef __attribute__((ext_vector_type(2))) float v2f;
typedef __attribute__((ext_vector_type(8))) float v8f;

#define VOCAB 50000
#define EMB   300
#define BATCH 256
#define SEQ   512
#define NCLS  10
#define NPROJ 16   // 12 projection columns padded to 16 (one 64B row per vocab entry)
#define NTILE (VOCAB / 16)  // 3125, exact

// ---------------------------------------------------------------------------
// Kernel 1: proj[v][n] = dot(emb_table[v,:], W_n)   (n = 12 conv-kernel rows)
// GEMM [50000,300] x [300,16] via V_WMMA_F32_16X16X4_F32, one wave per 16x16
// output tile, 8 waves/block sharing the 300x16 weight panel in LDS.
// ---------------------------------------------------------------------------
__global__ __launch_bounds__(256) void proj_gemm_wmma(
    const float* __restrict__ emb, const float* __restrict__ k3,
    const float* __restrict__ k4,  const float* __restrict__ k5,
    float* __restrict__ proj)
{
    __shared__ float wlds[EMB * NPROJ];   // [k][n], n-major, 19.2 KB
    const int tid = threadIdx.x;

    for (int i = tid; i < EMB * NPROJ; i += 256) {
        const int k = i >> 4, n = i & 15;
        float w = 0.0f;
        if      (n < 3)  w = k3[n * EMB + k];
        else if (n < 7)  w = k4[(n - 3) * EMB + k];
        else if (n < 12) w = k5[(n - 7) * EMB + k];
        wlds[i] = w;
    }
    __syncthreads();

    const int wave = tid >> 5;
    const int lane = tid & 31;
    const int tile = blockIdx.x * 8 + wave;
    if (tile >= NTILE) return;            // wave-uniform: EXEC all-ones inside

    const int row  = lane & 15;           // M (A) / N (B,D)
    const int half = lane >> 4;           // 0 -> K=0,1 ; 1 -> K=2,3
    const int koff = half * 2;
    const float* __restrict__ Arow = emb + (size_t)(tile * 16 + row) * EMB;

    v8f acc = {};
    for (int kb = 0; kb < EMB; kb += 4) {
        v2f a, b;
        a.x = Arow[kb + koff];
        a.y = Arow[kb + koff + 1];
        b.x = wlds[(kb + koff)     * NPROJ + row];
        b.y = wlds[(kb + koff + 1) * NPROJ + row];
        acc = __builtin_amdgcn_wmma_f32_16x16x4_f32(
            /*neg_a=*/false, a, /*neg_b=*/false, b,
            /*c_mod=*/(short)0, acc, /*reuse_a=*/false, /*reuse_b=*/false);
    }

    const int mbase = tile * 16 + half * 8;
    #pragma unroll
    for (int r = 0; r < 8; ++r)
        proj[(size_t)(mbase + r) * NPROJ + row] = acc[r];
}

// ---------------------------------------------------------------------------
// Kernel 2: per-sentence gather of proj rows, sliding-window sums,
// max-over-time (tanh deferred via monotonicity), FC + log_softmax.
// One block (8 waves) per sentence.
// ---------------------------------------------------------------------------
__global__ __launch_bounds__(256) void textcnn_tail(
    const int* __restrict__ sent, const float* __restrict__ proj,
    const float* __restrict__ b3, const float* __restrict__ b4,
    const float* __restrict__ b5, const float* __restrict__ fc_w,
    const float* __restrict__ fc_b, float* __restrict__ out)
{
    __shared__ float d[12][SEQ];          // 24 KB: per-token dot products
    __shared__ float red[3][256];
    const int b = blockIdx.x, tid = threadIdx.x;

    for (int t = tid; t < SEQ; t += 256) {
        const int v = sent[b * SEQ + t];
        const float4* pr = (const float4*)(proj + (size_t)v * NPROJ);
        const float4 p0 = pr[0], p1 = pr[1], p2 = pr[2];
        d[0][t] = p0.x; d[1][t] = p0.y; d[2][t]  = p0.z; d[3][t]  = p0.w;
        d[4][t] = p1.x; d[5][t] = p1.y; d[6][t]  = p1.z; d[7][t]  = p1.w;
        d[8][t] = p2.x; d[9][t] = p2.y; d[10][t] = p2.z; d[11][t] = p2.w;
    }
    __syncthreads();

    float m3 = -1e30f, m4 = -1e30f, m5 = -1e30f;
    for (int p = tid; p < SEQ - 2; p += 256)
        m3 = fmaxf(m3, d[0][p] + d[1][p + 1] + d[2][p + 2]);
    for (int p = tid; p < SEQ - 3; p += 256)
        m4 = fmaxf(m4, d[3][p] + d[4][p + 1] + d[5][p + 2] + d[6][p + 3]);
    for (int p = tid; p < SEQ - 4; p += 256)
        m5 = fmaxf(m5, d[7][p] + d[8][p + 1] + d[9][p + 2] + d[10][p + 3]
                        + d[11][p + 4]);

    red[0][tid] = m3; red[1][tid] = m4; red[2][tid] = m5;
    __syncthreads();
    for (int s = 128; s > 0; s >>= 1) {
        if (tid < s) {
            red[0][tid] = fmaxf(red[0][tid], red[0][tid + s]);
            red[1][tid] = fmaxf(red[1][tid], red[1][tid + s]);
            red[2][tid] = fmaxf(red[2][tid], red[2][tid + s]);
        }
        __syncthreads();
    }

    if (tid == 0) {
        const float f3 = tanhf(red[0][0] + b3[0]);
        const float f4 = tanhf(red[1][0] + b4[0]);
        const float f5 = tanhf(red[2][0] + b5[0]);
        float lg[NCLS], mx = -1e30f;
        #pragma unroll
        for (int c = 0; c < NCLS; ++c) {
            lg[c] = f3 * fc_w[c * 3 + 0] + f4 * fc_w[c * 3 + 1]
                  + f5 * fc_w[c * 3 + 2] + fc_b[c];
            mx = fmaxf(mx, lg[c]);
        }
        float se = 0.0f;
        #pragma unroll
        for (int c = 0; c < NCLS; ++c) se += expf(lg[c] - mx);
        const float lse = logf(se) + mx;
        #pragma unroll
        for (int c = 0; c < NCLS; ++c) out[b * NCLS + c] = lg[c] - lse;
    }
}

// ---------------------------------------------------------------------------
extern "C" void kernel_launch(void* const* d_in, const int* in_sizes, int n_in,
                              void* d_out, int out_size, void* d_ws, size_t ws_size,
                              hipStream_t stream)
{
    (void)in_sizes; (void)n_in; (void)out_size; (void)ws_size;
    const int*   sent = (const int*)  d_in[0];
    const float* emb  = (const float*)d_in[1];
    const float* k3   = (const float*)d_in[2];
    const float* b3   = (const float*)d_in[3];
    const float* k4   = (const float*)d_in[4];
    const float* b4   = (const float*)d_in[5];
    const float* k5   = (const float*)d_in[6];
    const float* b5   = (const float*)d_in[7];
    const float* fcw  = (const float*)d_in[8];
    const float* fcb  = (const float*)d_in[9];

    float* proj = (float*)d_ws;           // VOCAB * 16 * 4 B = 3.2 MB scratch

    proj_gemm_wmma<<<(NTILE + 7) / 8, 256, 0, stream>>>(emb, k3, k4, k5, proj);
    textcnn_tail<<<BATCH, 256, 0, stream>>>(sent, proj, b3, b4, b5, fcw, fcb,
                                            (float*)d_out);
}